// Attention_71347996721755
// MI455X (gfx1250) — compile-verified
//
#include <hip/hip_runtime.h>
#include <hip/hip_bf16.h>

// ---------------------------------------------------------------------------
// Problem constants (B,C,H,W) = (16,64,64,64)
// ---------------------------------------------------------------------------
#define BB   16
#define CC   64
#define HH   64
#define WW   64
#define HWN  4096      // H*W
#define HWP  1024      // pooled (H/2)*(W/2)
#define CQ   8         // C/8  (q,k channels)
#define CV   32        // C/2  (v channels)
#define NEG_BIG (-3.0e38f)

typedef __attribute__((ext_vector_type(2))) float v2f;
typedef __attribute__((ext_vector_type(8))) float v8f;

static __device__ __forceinline__ v8f wmma_f32(v2f a, v2f b, v8f c) {
    // V_WMMA_F32_16X16X4_F32 : D(16x16,f32) = A(16x4,f32) x B(4x16,f32) + C
    return __builtin_amdgcn_wmma_f32_16x16x4_f32(
        /*neg_a=*/false, a, /*neg_b=*/false, b,
        /*c_mod=*/(short)0, c, /*reuse_a=*/false, /*reuse_b=*/false);
}

// ---------------------------------------------------------------------------
// VALU-only butterfly reductions across each 16-lane half via
// v_permlane16_b32 (immediate lane-select; no LDS, no DScnt stalls).
// Select nibbles encode lane^1 / lane^2 / lane^4 / lane^8 within each half.
// ---------------------------------------------------------------------------
static __device__ __forceinline__ float xor16(float v, unsigned lo, unsigned hi) {
    unsigned s = __builtin_bit_cast(unsigned, v);
    unsigned r = (unsigned)__builtin_amdgcn_permlane16(
        (int)s, (int)s, (int)lo, (int)hi, false, false);
    return __builtin_bit_cast(float, r);
}
static __device__ __forceinline__ float redmax16(float v) {
    v = fmaxf(v, xor16(v, 0x67452301u, 0xEFCDAB89u)); // ^1
    v = fmaxf(v, xor16(v, 0x54761032u, 0xDCFE98BAu)); // ^2
    v = fmaxf(v, xor16(v, 0x32107654u, 0xBA98FEDCu)); // ^4
    v = fmaxf(v, xor16(v, 0xFEDCBA98u, 0x76543210u)); // ^8
    return v;
}
static __device__ __forceinline__ float redsum16(float v) {
    v += xor16(v, 0x67452301u, 0xEFCDAB89u); // ^1
    v += xor16(v, 0x54761032u, 0xDCFE98BAu); // ^2
    v += xor16(v, 0x32107654u, 0xBA98FEDCu); // ^4
    v += xor16(v, 0xFEDCBA98u, 0x76543210u); // ^8
    return v;
}

// ---------------------------------------------------------------------------
// Kernel 1: q = Wq*x ; k = maxpool2(Wk*x) ; v = maxpool2(Wv*x)
//   Qt layout: [b][n][8]      (fragment-friendly: 8 contiguous c per query)
//   Kt layout: [b][m][8]
//   Vb layout: [b][vc][1024]  (key index contiguous -> float2 B-fragments)
// One thread per pooled pixel (handles its 2x2 block of full-res pixels).
// ---------------------------------------------------------------------------
__global__ __launch_bounds__(256) void qkv_kernel(
    const float* __restrict__ x,
    const float* __restrict__ Wq, const float* __restrict__ Wk,
    const float* __restrict__ Wv,
    float* __restrict__ Qt, float* __restrict__ Kt, float* __restrict__ Vb)
{
    __shared__ float wq[CQ * CC];
    __shared__ float wk[CQ * CC];
    __shared__ float wv[CV * CC];

    const int tid = threadIdx.x;
    for (int i = tid; i < CQ * CC; i += 256) { wq[i] = Wq[i]; wk[i] = Wk[i]; }
    for (int i = tid; i < CV * CC; i += 256) { wv[i] = Wv[i]; }
    __syncthreads();

    const int gid = blockIdx.x * 256 + tid;     // B*HWP threads total
    const int b   = gid >> 10;                  // / 1024
    const int mp  = gid & (HWP - 1);            // pooled pixel index
    const int hp  = mp >> 5;
    const int wp  = mp & 31;

    float kmax[CQ], vmax[CV];
#pragma unroll
    for (int o = 0; o < CQ; ++o) kmax[o] = NEG_BIG;
#pragma unroll
    for (int o = 0; o < CV; ++o) vmax[o] = NEG_BIG;

    const float* xb = x + (size_t)b * CC * HWN;

#pragma unroll 1
    for (int p = 0; p < 4; ++p) {
        const int h = hp * 2 + (p >> 1);
        const int w = wp * 2 + (p & 1);
        const int n = h * WW + w;

        float qa[CQ], ka[CQ], va[CV];
#pragma unroll
        for (int o = 0; o < CQ; ++o) { qa[o] = 0.f; ka[o] = 0.f; }
#pragma unroll
        for (int o = 0; o < CV; ++o) { va[o] = 0.f; }

        const float* xp = xb + n;
#pragma unroll 4
        for (int c = 0; c < CC; ++c) {
            const float xv = xp[(size_t)c * HWN];
#pragma unroll
            for (int o = 0; o < CQ; ++o) {
                qa[o] = fmaf(wq[o * CC + c], xv, qa[o]);
                ka[o] = fmaf(wk[o * CC + c], xv, ka[o]);
            }
#pragma unroll
            for (int o = 0; o < CV; ++o)
                va[o] = fmaf(wv[o * CC + c], xv, va[o]);
        }

        float* qo = Qt + ((size_t)b * HWN + n) * CQ;
#pragma unroll
        for (int o = 0; o < CQ; ++o) {
            qo[o]   = qa[o];
            kmax[o] = fmaxf(kmax[o], ka[o]);
        }
#pragma unroll
        for (int o = 0; o < CV; ++o) vmax[o] = fmaxf(vmax[o], va[o]);
    }

    float* ko = Kt + ((size_t)b * HWP + mp) * CQ;
#pragma unroll
    for (int o = 0; o < CQ; ++o) ko[o] = kmax[o];

    float* vo = Vb + (size_t)b * CV * HWP + mp;
#pragma unroll
    for (int o = 0; o < CV; ++o) vo[(size_t)o * HWP] = vmax[o];
}

// ---------------------------------------------------------------------------
// Kernel 2: streaming-softmax attention, f32 WMMA 16x16x4.
//  Block = 256 threads = 8 waves; block owns 128 query rows of one batch,
//  each wave owns 16 rows. Keys streamed in chunks of 64; scores never
//  touch global memory. P is transposed through a per-wave LDS slab
//  (row stride 68 floats -> 4-bank skew, conflict-free b64 reads).
//  Softmax row-stats reduced with v_permlane16 (pure VALU cross-lane).
// ---------------------------------------------------------------------------
#define MCHUNK 64
#define PSTRIDE 68   // 64 + 4 pad

__global__ __launch_bounds__(256) void attn_kernel(
    const float* __restrict__ Qt, const float* __restrict__ Kt,
    const float* __restrict__ Vb, float* __restrict__ Ob)
{
    __shared__ float plds[8 * 16 * PSTRIDE];   // 34816 B

    const int tid   = threadIdx.x;
    const int wave  = tid >> 5;
    const int lane  = tid & 31;
    const int lh    = lane & 15;       // N / M index within fragment
    const int half  = lane >> 4;       // selects K-pair {0,1} vs {2,3}
    const int kpair = half * 2;

    const int b     = blockIdx.x >> 5;          // / (HWN/128)
    const int qtile = blockIdx.x & 31;
    const int n0    = qtile * 128 + wave * 16;  // this wave's query base
    const int qrow  = n0 + lh;

    const float* Qb  = Qt + (size_t)b * HWN * CQ;
    const float* Kb  = Kt + (size_t)b * HWP * CQ;
    const float* Vbb = Vb + (size_t)b * CV * HWP;

    // A-fragments of Qt (reused for every key chunk): lane l holds row lh,
    // K elements {kpair, kpair+1} of c-groups [0..3] and [4..7].
    const float* qp = Qb + (size_t)qrow * CQ;
    const v2f qa0 = *(const v2f*)(qp + kpair);
    const v2f qa1 = *(const v2f*)(qp + 4 + kpair);

    float rowmax[8], rowsum[8];
#pragma unroll
    for (int r = 0; r < 8; ++r) { rowmax[r] = NEG_BIG; rowsum[r] = 0.f; }
    v8f o0 = {}; v8f o1 = {};

    float* pl = plds + wave * 16 * PSTRIDE;

#pragma unroll 1
    for (int mc = 0; mc < HWP; mc += MCHUNK) {
        // ---- warm L2/WGP$ for the next chunk while this one computes ----
        {
            const int nmc = (mc + MCHUNK) & (HWP - 1);
            // next K chunk: 64 keys x 8 f32 = 2KB contiguous
            __builtin_prefetch(Kb + (size_t)nmc * CQ + lane * 16, 0, 3);
            // next V chunk: one 256B row slice per lane (32 vc rows)
            __builtin_prefetch(Vbb + (size_t)lane * HWP + nmc, 0, 3);
        }

        // ---- scores: 4 tiles of 16x16, K-depth = 8 channels (2 WMMAs) ----
        v8f s[4];
#pragma unroll
        for (int j = 0; j < 4; ++j) {
            const int mcol = mc + j * 16 + lh;
            const float* kp = Kb + (size_t)mcol * CQ;
            const v2f kb0 = *(const v2f*)(kp + kpair);
            const v2f kb1 = *(const v2f*)(kp + 4 + kpair);
            v8f acc = {};
            acc  = wmma_f32(qa0, kb0, acc);
            s[j] = wmma_f32(qa1, kb1, acc);
        }

        // ---- online softmax (row r lives in one 16-lane half) ----
#pragma unroll
        for (int r = 0; r < 8; ++r) {
            float m = fmaxf(fmaxf(s[0][r], s[1][r]), fmaxf(s[2][r], s[3][r]));
            m = redmax16(m);
            const float newm = fmaxf(rowmax[r], m);
            const float corr = __expf(rowmax[r] - newm);
            rowmax[r] = newm;
            float lsum = 0.f;
#pragma unroll
            for (int j = 0; j < 4; ++j) {
                const float pv = __expf(s[j][r] - newm);
                s[j][r] = pv;
                lsum += pv;
            }
            lsum = redsum16(lsum);
            rowsum[r] = rowsum[r] * corr + lsum;
            o0[r] *= corr;
            o1[r] *= corr;
        }

        // ---- transpose P through this wave's private LDS slab ----
#pragma unroll
        for (int j = 0; j < 4; ++j)
#pragma unroll
            for (int r = 0; r < 8; ++r)
                pl[(r + half * 8) * PSTRIDE + j * 16 + lh] = s[j][r];
        // same-wave store->load ordering; no barrier needed, just DScnt
        asm volatile("s_wait_dscnt 0" ::: "memory");

        // ---- O += P(16xMCHUNK) x V^T(MCHUNKx32): 16 K-steps, 2 tiles ----
#pragma unroll 4
        for (int s4 = 0; s4 < MCHUNK / 4; ++s4) {
            const v2f pa = *(const v2f*)(pl + lh * PSTRIDE + s4 * 4 + kpair);
            const int gm = mc + s4 * 4 + kpair;
            const v2f vb0 = *(const v2f*)(Vbb + (size_t)lh        * HWP + gm);
            const v2f vb1 = *(const v2f*)(Vbb + (size_t)(16 + lh) * HWP + gm);
            o0 = wmma_f32(pa, vb0, o0);
            o1 = wmma_f32(pa, vb1, o1);
        }
    }

    // ---- normalize and write O[b][vc][n] ----
    float* Obp = Ob + (size_t)b * CV * HWN;
#pragma unroll
    for (int r = 0; r < 8; ++r) {
        const float inv = 1.0f / rowsum[r];
        const int n = n0 + r + half * 8;
        Obp[(size_t)lh        * HWN + n] = o0[r] * inv;
        Obp[(size_t)(16 + lh) * HWN + n] = o1[r] * inv;
    }
}

// ---------------------------------------------------------------------------
// Kernel 3: out = gamma * (Wo @ O) + x    (one thread per pixel)
// ---------------------------------------------------------------------------
__global__ __launch_bounds__(256) void proj_kernel(
    const float* __restrict__ Ob, const float* __restrict__ Wo,
    const float* __restrict__ x, const float* __restrict__ gamma,
    float* __restrict__ out)
{
    __shared__ float wo[CC * CV];
    const int tid = threadIdx.x;
    for (int i = tid; i < CC * CV; i += 256) wo[i] = Wo[i];
    __syncthreads();

    const int gid = blockIdx.x * 256 + tid;
    const int b   = gid >> 12;          // / 4096
    const int n   = gid & (HWN - 1);

    const float* op = Ob + (size_t)b * CV * HWN + n;
    float ov[CV];
#pragma unroll
    for (int vc = 0; vc < CV; ++vc) ov[vc] = op[(size_t)vc * HWN];

    const float g = gamma[0];
    const float* xp = x   + (size_t)b * CC * HWN + n;
    float*       yp = out + (size_t)b * CC * HWN + n;
#pragma unroll 4
    for (int co = 0; co < CC; ++co) {
        float acc = 0.f;
#pragma unroll
        for (int vc = 0; vc < CV; ++vc)
            acc = fmaf(wo[co * CV + vc], ov[vc], acc);
        yp[(size_t)co * HWN] = g * acc + xp[(size_t)co * HWN];
    }
}

// ---------------------------------------------------------------------------
extern "C" void kernel_launch(void* const* d_in, const int* in_sizes, int n_in,
                              void* d_out, int out_size, void* d_ws, size_t ws_size,
                              hipStream_t stream) {
    (void)in_sizes; (void)n_in; (void)out_size; (void)ws_size;
    const float* x     = (const float*)d_in[0];
    const float* Wq    = (const float*)d_in[1];
    const float* Wk    = (const float*)d_in[2];
    const float* Wv    = (const float*)d_in[3];
    const float* Wo    = (const float*)d_in[4];
    const float* gamma = (const float*)d_in[5];
    float* out = (float*)d_out;

    float* ws = (float*)d_ws;
    float* Qt = ws;                                     // B*HWN*CQ  = 524288 f
    float* Kt = Qt + (size_t)BB * HWN * CQ;             // B*HWP*CQ  = 131072 f
    float* Vb = Kt + (size_t)BB * HWP * CQ;             // B*CV*HWP  = 524288 f
    float* Ob = Vb + (size_t)BB * CV * HWP;             // B*CV*HWN  = 2097152 f
                                                        // total 13.1 MB

    qkv_kernel <<<BB * HWP / 256,        256, 0, stream>>>(x, Wq, Wk, Wv, Qt, Kt, Vb);
    attn_kernel<<<BB * (HWN / 128),      256, 0, stream>>>(Qt, Kt, Vb, Ob);
    proj_kernel<<<BB * HWN / 256,        256, 0, stream>>>(Ob, Wo, x, gamma, out);
}